// IrrepsStrideLinear_74723841016348
// MI455X (gfx1250) — compile-verified
//
#include <hip/hip_runtime.h>
#include <stdint.h>

typedef float v2f __attribute__((ext_vector_type(2)));
typedef float v4f __attribute__((ext_vector_type(4)));
typedef float v8f __attribute__((ext_vector_type(8)));
typedef unsigned int v4u __attribute__((ext_vector_type(4)));
typedef int v4i __attribute__((ext_vector_type(4)));
typedef int v8i __attribute__((ext_vector_type(8)));

#define MUL   128          // mul_in == mul_out
#define IRD   16           // (LMAX+1)^2 spherical channels
#define NPG   16           // nodes per workgroup
#define NSTR  20           // lds_x per-node stride (16 c + 4 pad dwords, TDM pad)
#define XROW  (NPG * NSTR) // 320 dwords per i-row
#define OROW  17           // lds_o per-o stride (16 c + 1 pad) -> conflict-free scatter
#define ONODE (128 * OROW) // lds_o per-node stride (2176 dwords)

// D = A(16x4: nodes x i) * B(4x16: i x o) + C, accumulated over 32 K-steps.
// wave w owns o-tile [16w, 16w+16); 16 channel-tiles (one per c) per 16-node group.
__global__ __launch_bounds__(256, 1)
void irreps_stride_linear_wmma(const float* __restrict__ xg_all,
                               const float* __restrict__ w0,
                               const float* __restrict__ w1,
                               const float* __restrict__ w2,
                               const float* __restrict__ w3,
                               float* __restrict__ out_all)
{
    __shared__ float lds_x[128 * XROW + 8]; // 160 KB: x block, [i][node*20 + c] (TDM-padded)
    __shared__ float lds_o[NPG * ONODE];    // 136 KB: out block, [node][o*17 + c]

    const int tid  = threadIdx.x;
    const int wave = tid >> 5;     // 0..7 -> o-tile
    const int lane = tid & 31;
    const int m    = lane & 15;    // A: node row / B,D: o column
    const int h    = lane >> 4;    // half-wave selector (K split / M+8 split)
    const int o0   = wave << 4;

    const long grp = blockIdx.x;
    const float* xg = xg_all  + grp * (long)(NPG * MUL * IRD);
    float*       og = out_all + grp * (long)(NPG * MUL * IRD);

#if __has_builtin(__builtin_amdgcn_tensor_load_to_lds)
    // ---- TDM load: one DMA stages + transposes + pads the x block ----
    // dim0 = c (16 contiguous), dim1 = n (16, stride 2048), dim2 = i (128, stride 16)
    // LDS gets [i][n][c] with 4 pad dwords after every 16-dword line -> node stride 20.
    if (wave == 0) {
        const unsigned long long ga = (unsigned long long)(uintptr_t)xg;
        const unsigned int ldsb = (unsigned int)(uintptr_t)(void*)lds_x; // LDS byte offset
        v4u g0;
        g0[0] = 1u;                                        // count=1, user descriptor
        g0[1] = ldsb;                                      // lds_addr (bytes)
        g0[2] = (unsigned int)(ga & 0xffffffffu);          // global_addr[31:0]
        g0[3] = ((unsigned int)(ga >> 32) & 0x01ffffffu)   // global_addr[56:32]
                | (2u << 30);                              // type=2 ("image")
        v8i g1;
        g1[0] = (int)((2u << 16)      // data_size = 4B
                    | (1u << 20)      // pad_enable
                    | (3u << 22)      // pad_interval: 2^(3+1) = 16 dwords
                    | (3u << 25));    // pad_amount: 3+1 = 4 dwords
        g1[1] = (int)(16u << 16);     // tensor_dim0 = 16 (c)
        g1[2] = (int)(16u << 16);     // tensor_dim1 = 16 (n)
        g1[3] = (int)(16u << 16);     // tile_dim0 = 16
        g1[4] = (int)(16u | (128u << 16)); // tile_dim1 = 16 (n), tile_dim2 = 128 (i)
        g1[5] = 2048;                 // tensor_dim0_stride = 2048 (node stride, elems)
        g1[6] = (int)(16u << 16);     // tensor_dim1_stride = 16 (i stride, elems)
        g1[7] = 0;
        v4i g2; g2[0] = 128; g2[1] = 0; g2[2] = 0; g2[3] = 0; // tensor_dim2 = 128 (i)
        v4i g3; g3[0] = 0;  g3[1] = 0;  g3[2] = 0;  g3[3] = 0;
        v8i g4; g4[0] = 0;  g4[1] = 0;  g4[2] = 0;  g4[3] = 0;
                g4[4] = 0;  g4[5] = 0;  g4[6] = 0;  g4[7] = 0;
        __builtin_amdgcn_tensor_load_to_lds(g0, g1, g2, g3, g4, 0);
        __builtin_amdgcn_s_wait_tensorcnt(0);
    }
#else
    // ---- Fallback: manual staged copy (coalesced b128 loads + LDS stores) ----
    #pragma unroll 4
    for (int it = 0; it < 32; ++it) {
        int f4 = it * 256 + tid;
        int f  = f4 << 2;              // flat: n*2048 + i*16 + c
        int n  = f >> 11;
        int i  = (f >> 4) & 127;
        int c  = f & 15;
        v4f v = *(const v4f*)(xg + f);
        *(v4f*)(&lds_x[i * XROW + n * NSTR + c]) = v;
    }
#endif
    __syncthreads();

    #pragma unroll 1
    for (int l = 0; l < 4; ++l) {
        const float* w = (l == 0) ? w0 : (l == 1) ? w1 : (l == 2) ? w2 : w3;
        const int c0 = l * l;          // first channel of degree l
        const int cn = 2 * l + 1;      // channels of degree l

        // B fragments: B[k,n] = w[i_k, o0+n]; VGPR r, half h -> K = r + 2h
        float breg[64];
        #pragma unroll
        for (int s = 0; s < 32; ++s) {
            const int i0 = 4 * s + 2 * h;
            breg[2 * s]     = w[(i0    ) * MUL + o0 + m];
            breg[2 * s + 1] = w[(i0 + 1) * MUL + o0 + m];
        }

        for (int cc = 0; cc < cn; ++cc) {
            const int c = c0 + cc;
            v8f acc = {0.f, 0.f, 0.f, 0.f, 0.f, 0.f, 0.f, 0.f};
            #pragma unroll
            for (int s = 0; s < 32; ++s) {
                const int i0 = 4 * s + 2 * h;
                v2f a;  // A[m,k] = x[node_m, i_k, c] from padded LDS
                a.x = lds_x[(i0    ) * XROW + m * NSTR + c];
                a.y = lds_x[(i0 + 1) * XROW + m * NSTR + c];
                v2f b;
                b.x = breg[2 * s];
                b.y = breg[2 * s + 1];
                acc = __builtin_amdgcn_wmma_f32_16x16x4_f32(
                        false, a, false, b, (short)0, acc, false, false);
            }
            // D scatter: lane -> o = o0+m ; VGPR r -> node = r + 8h
            #pragma unroll
            for (int r = 0; r < 8; ++r) {
                const int node = r + 8 * h;
                lds_o[node * ONODE + (o0 + m) * OROW + c] = acc[r];
            }
        }
    }
    __syncthreads();

#if __has_builtin(__builtin_amdgcn_tensor_store_from_lds)
    // ---- TDM store: one DMA de-stages the padded out block ----
    // LDS walk is linear over lds_o = [n][o*17 + c]. Tile line = 17 elements but
    // tensor_dim0 = 16, so the per-line pad dword is dropped by the per-dim OOB rule.
    // dim1 = o (128 lines, stride 16), dim2 = n (16, stride 2048) -> dense global image.
    if (wave == 0) {
        const unsigned long long ga = (unsigned long long)(uintptr_t)og;
        const unsigned int ldsb = (unsigned int)(uintptr_t)(void*)lds_o;
        v4u g0;
        g0[0] = 1u;                                        // count=1, user descriptor
        g0[1] = ldsb;                                      // lds_addr (bytes)
        g0[2] = (unsigned int)(ga & 0xffffffffu);          // global_addr[31:0]
        g0[3] = ((unsigned int)(ga >> 32) & 0x01ffffffu)   // global_addr[56:32]
                | (2u << 30);                              // type=2 ("image")
        v8i g1;
        g1[0] = (int)(2u << 16);      // data_size = 4B, no padding on store
        g1[1] = (int)(16u << 16);     // tensor_dim0 = 16 (c) -> drops 17th lane of tile
        g1[2] = (int)(128u << 16);    // tensor_dim1 = 128 (o)
        g1[3] = (int)(17u << 16);     // tile_dim0 = 17 (16 c + 1 LDS pad dword)
        g1[4] = (int)(128u | (16u << 16)); // tile_dim1 = 128 (o), tile_dim2 = 16 (n)
        g1[5] = 16;                   // tensor_dim0_stride = 16 (o stride, elems)
        g1[6] = (int)(2048u << 16);   // tensor_dim1_stride = 2048 (n stride, elems)
        g1[7] = 0;
        v4i g2; g2[0] = 16; g2[1] = 0; g2[2] = 0; g2[3] = 0;  // tensor_dim2 = 16 (n)
        v4i g3; g3[0] = 0;  g3[1] = 0;  g3[2] = 0;  g3[3] = 0;
        v8i g4; g4[0] = 0;  g4[1] = 0;  g4[2] = 0;  g4[3] = 0;
                g4[4] = 0;  g4[5] = 0;  g4[6] = 0;  g4[7] = 0;
        __builtin_amdgcn_tensor_store_from_lds(g0, g1, g2, g3, g4, 0);
        __builtin_amdgcn_s_wait_tensorcnt(0);  // LDS must stay valid until DMA drains
    }
#else
    // ---- Fallback: coalesced manual write-out: LDS (padded) -> global (dense) ----
    #pragma unroll 4
    for (int it = 0; it < 128; ++it) {
        int f = it * 256 + tid;        // n*2048 + o*16 + c
        int n = f >> 11;
        int o = (f >> 4) & 127;
        int c = f & 15;
        og[f] = lds_o[n * ONODE + o * OROW + c];
    }
#endif
}

extern "C" void kernel_launch(void* const* d_in, const int* in_sizes, int n_in,
                              void* d_out, int out_size, void* d_ws, size_t ws_size,
                              hipStream_t stream) {
    const float* x  = (const float*)d_in[0];
    const float* w0 = (const float*)d_in[1];
    const float* w1 = (const float*)d_in[2];
    const float* w2 = (const float*)d_in[3];
    const float* w3 = (const float*)d_in[4];
    float* out = (float*)d_out;

    const int N      = in_sizes[0] / (MUL * IRD);  // 100000
    const int groups = N / NPG;                    // 6250 (exact)
    irreps_stride_linear_wmma<<<dim3(groups), dim3(256), 0, stream>>>(
        x, w0, w1, w2, w3, out);
}